// Decoder_38182259261791
// MI455X (gfx1250) — compile-verified
//
#include <hip/hip_runtime.h>
#include <hip/hip_bf16.h>

// ---------------------------------------------------------------------------
// NetVLAD-style head for MI455X (gfx1250, wave32).
// Heavy math via V_WMMA_F32_16X16X4_F32 (fp32-preserving matrix path).
// LDS tiles stored so every WMMA fragment is one aligned ds_load_b64.
// Register-staged double buffering overlaps HBM loads with the WMMA stream.
// ---------------------------------------------------------------------------

typedef __attribute__((ext_vector_type(2))) float v2f;
typedef __attribute__((ext_vector_type(8))) float v8f;

#define KCLU   8
#define NCLU   10          // K_CLUSTERS + G_CLUSTERS
#define MID    512
#define OUTD   5994
#define BATCH  32
#define WIDTH  500
#define KDIM   (MID * 7)   // 3584

// ---------------------------------------------------------------------------
// Generic LDS-tiled fp32 WMMA GEMM:  C(MxN) = op(A(MxK) @ B(KxN)) [+bias][relu]
//   A row-major (lda).  B row-major (ldb) or, if TRANS_B, element (k,n) is at
//   B[n*ldb + k].  BIAS_MODE: 0=none, 1=per-M row, 2=per-N col.
//   Batched over blockIdx.z with element strides.
// Block: 256 threads = 8 wave32, tile 64(M) x 128(N), K-step 16.
// Waves in a 4(M) x 2(N) grid; each wave owns a 16x64 strip = 4 accumulators.
// LDS: As[64][20] (row-major MxK), Bs[128][20] (transposed: [n][k]).
//   Stride 20 dwords => 16B-aligned b128 stores, 8B-aligned b64 frag reads,
//   and 20*n mod 64 distinct for n in [0,16) => conflict-free frag reads.
// ---------------------------------------------------------------------------
template <bool TRANS_B, int BIAS_MODE, bool RELU>
__global__ __launch_bounds__(256) void gemm_wmma_f32(
    const float* __restrict__ A, const float* __restrict__ B,
    const float* __restrict__ bias, float* __restrict__ C,
    int M, int N, int K, int lda, int ldb, int ldc,
    long strideA, long strideB, long strideC)
{
    constexpr int MB = 64, NB = 128, KB = 16, KP = KB + 4;
    __shared__ float As[MB][KP];
    __shared__ float Bs[NB][KP];   // transposed: Bs[n][k]

    const int bz = blockIdx.z;
    A += (long)bz * strideA;
    B += (long)bz * strideB;
    C += (long)bz * strideC;

    const int tid  = threadIdx.x;
    const int lane = tid & 31;
    const int wave = tid >> 5;
    const int wm   = wave >> 1;   // 0..3
    const int wn   = wave & 1;    // 0..1

    const int block_m = blockIdx.y * MB;
    const int block_n = blockIdx.x * NB;

    // A-frag addressing (ISA 16x4 f32 A layout): lanes 0-15 -> K={0,1},
    // lanes 16-31 -> K={2,3}; lane&15 selects the M row / N col.
    const int frag_mn = lane & 15;
    const int frag_k  = (lane >> 4) << 1;   // 0 or 2

    // cooperative-load assignments (fixed per thread)
    const int arow = tid >> 2;          // 0..63
    const int acol = (tid & 3) * 4;     // 0,4,8,12

    // ---- tile loaders: global -> registers (guarded, widest safe vector) ---
    auto load_a = [&](int k0, float4& r) {
        const int gm = block_m + arow;
        const int gk = k0 + acol;
        if (gm < M && gk + 3 < K) {
            r = *(const float4*)(A + (long)gm * lda + gk);   // lda % 4 == 0
        } else {
            r.x = (gm < M && gk + 0 < K) ? A[(long)gm * lda + gk + 0] : 0.f;
            r.y = (gm < M && gk + 1 < K) ? A[(long)gm * lda + gk + 1] : 0.f;
            r.z = (gm < M && gk + 2 < K) ? A[(long)gm * lda + gk + 2] : 0.f;
            r.w = (gm < M && gk + 3 < K) ? A[(long)gm * lda + gk + 3] : 0.f;
        }
    };
    auto load_b = [&](int k0, float2 (&r)[4]) {
        #pragma unroll
        for (int i = 0; i < 4; ++i) {
            const int v = tid + i * 256;          // 0..1023 float2 slots
            float2 t;
            if (!TRANS_B) {
                const int row = v >> 6;           // K: 64 float2 per row
                const int c2  = (v & 63) * 2;     // N
                const int gk = k0 + row, gn = block_n + c2;
                if (gk < K && gn + 1 < N) {       // ldb even => 8B aligned
                    t = *(const float2*)(B + (long)gk * ldb + gn);
                } else {
                    t.x = (gk < K && gn + 0 < N) ? B[(long)gk * ldb + gn + 0] : 0.f;
                    t.y = (gk < K && gn + 1 < N) ? B[(long)gk * ldb + gn + 1] : 0.f;
                }
            } else {
                const int nl = v >> 3;            // N: 8 float2 per n row
                const int c2 = (v & 7) * 2;       // K
                const int gn = block_n + nl, gk = k0 + c2;
                if (gn < N && gk + 1 < K) {
                    t = *(const float2*)(B + (long)gn * ldb + gk);
                } else {
                    t.x = (gn < N && gk + 0 < K) ? B[(long)gn * ldb + gk + 0] : 0.f;
                    t.y = (gn < N && gk + 1 < K) ? B[(long)gn * ldb + gk + 1] : 0.f;
                }
            }
            r[i] = t;
        }
    };
    // ---- registers -> LDS ----
    auto store_a = [&](const float4& r) {
        *(float4*)&As[arow][acol] = r;            // 16B-aligned ds_store_b128
    };
    auto store_b = [&](const float2 (&r)[4]) {
        #pragma unroll
        for (int i = 0; i < 4; ++i) {
            const int v = tid + i * 256;
            if (!TRANS_B) {
                const int row = v >> 6;
                const int c2  = (v & 63) * 2;
                Bs[c2 + 0][row] = r[i].x;         // transpose on store
                Bs[c2 + 1][row] = r[i].y;
            } else {
                const int nl = v >> 3;
                const int c2 = (v & 7) * 2;
                *(float2*)&Bs[nl][c2] = r[i];     // already [n][k]
            }
        }
    };

    v8f acc[4];
    #pragma unroll
    for (int j = 0; j < 4; ++j)
        acc[j] = (v8f){0.f, 0.f, 0.f, 0.f, 0.f, 0.f, 0.f, 0.f};

    // ---- prologue: tile 0 ----
    float4 ar;  float2 br[4];
    load_a(0, ar); load_b(0, br);
    store_a(ar);   store_b(br);
    __syncthreads();

    for (int k0 = 0; k0 < K; k0 += KB) {
        const bool has_next = (k0 + KB) < K;

        // issue next tile's global loads before compute (latency overlap)
        float4 ar2;  float2 br2[4];
        if (has_next) {
            load_a(k0 + KB, ar2);
            load_b(k0 + KB, br2);
            // keep one more tile in flight in GL2 (global_prefetch_b8)
            if (k0 + 2 * KB < K) {
                const int pm = block_m + arow;
                if (pm < M)
                    __builtin_prefetch(&A[(long)pm * lda + k0 + 2 * KB], 0, 1);
            }
        }

        // ---- 4 K-steps of 4, 4 N-tiles: 16x V_WMMA_F32_16X16X4_F32 ----
        #pragma unroll
        for (int kk = 0; kk < KB; kk += 4) {
            const int ka = kk + frag_k;                       // even
            const v2f af = *(const v2f*)&As[wm * 16 + frag_mn][ka];
            #pragma unroll
            for (int j = 0; j < 4; ++j) {
                const int nl = wn * 64 + j * 16 + frag_mn;
                const v2f bf = *(const v2f*)&Bs[nl][ka];      // ds_load_b64
                acc[j] = __builtin_amdgcn_wmma_f32_16x16x4_f32(
                    false, af, false, bf, (short)0, acc[j], false, false);
            }
        }
        __syncthreads();                 // LDS fully consumed

        if (has_next) {
            store_a(ar2); store_b(br2);  // commit next tile
            __syncthreads();
        }
    }

    // ---- epilogue: C/D layout -> VGPR r holds row r (lanes 0-15) / r+8 ----
    #pragma unroll
    for (int j = 0; j < 4; ++j) {
        const int n = block_n + wn * 64 + j * 16 + frag_mn;
        #pragma unroll
        for (int r = 0; r < 8; ++r) {
            const int m = block_m + wm * 16 + (lane >> 4) * 8 + r;
            if (m < M && n < N) {
                float v = acc[j][r];
                if (BIAS_MODE == 1) v += bias[m];
                if (BIAS_MODE == 2) v += bias[n];
                if (RELU) v = fmaxf(v, 0.f);
                C[(long)m * ldc + n] = v;
            }
        }
    }
}

// ---------------------------------------------------------------------------
// softmax over the 10 cluster scores for each (b, w)
// ---------------------------------------------------------------------------
__global__ void softmax_k(const float* __restrict__ score,
                          float* __restrict__ soft)
{
    const int idx = blockIdx.x * blockDim.x + threadIdx.x;
    if (idx >= BATCH * WIDTH) return;
    const int b = idx / WIDTH, w = idx - b * WIDTH;
    const float* s = score + (long)b * NCLU * WIDTH + w;
    float v[NCLU];
    float mx = -INFINITY;
    #pragma unroll
    for (int k = 0; k < NCLU; ++k) { v[k] = s[k * WIDTH]; mx = fmaxf(mx, v[k]); }
    float sum = 0.f;
    #pragma unroll
    for (int k = 0; k < NCLU; ++k) { v[k] = __expf(v[k] - mx); sum += v[k]; }
    const float inv = 1.f / sum;
    float* o = soft + (long)b * NCLU * WIDTH + w;
    #pragma unroll
    for (int k = 0; k < NCLU; ++k) o[k * WIDTH] = v[k] * inv;
}

// sa_sum[b,k] = sum_w soft[b,k,w]   (320 rows of 500 -- trivial)
__global__ void sasum_k(const float* __restrict__ soft, float* __restrict__ sa)
{
    const int idx = blockIdx.x * blockDim.x + threadIdx.x;
    if (idx >= BATCH * NCLU) return;
    const float* s = soft + (long)idx * WIDTH;
    float t = 0.f;
    for (int w = 0; w < WIDTH; ++w) t += s[w];
    sa[idx] = t;
}

// residual vs centroid + intra-normalization, one block per (b, k<8)
__global__ __launch_bounds__(256) void cluster_k(
    const float* __restrict__ agg, const float* __restrict__ sa,
    const float* __restrict__ centroids, float* __restrict__ emb0)
{
    const int b = blockIdx.x / KCLU, k = blockIdx.x % KCLU;
    const float* a  = agg + ((long)b * NCLU + k) * MID;
    const float* ct = centroids + (long)k * MID;
    const float  s  = sa[b * NCLU + k];
    const int t = threadIdx.x;
    const float r0 = a[t]       - s * ct[t];
    const float r1 = a[t + 256] - s * ct[t + 256];
    __shared__ float red[256];
    red[t] = r0 * r0 + r1 * r1;
    __syncthreads();
    for (int off = 128; off > 0; off >>= 1) {
        if (t < off) red[t] += red[t + off];
        __syncthreads();
    }
    const float inv = 1.f / fmaxf(sqrtf(red[0]), 1e-12f);
    float* o = emb0 + ((long)b * KCLU + k) * MID;
    o[t]       = r0 * inv;
    o[t + 256] = r1 * inv;
}

// batch-norm over B=32 (population var) + relu; one thread per feature
__global__ void bn_relu_k(const float* __restrict__ h,
                          const float* __restrict__ gamma,
                          const float* __restrict__ beta,
                          float* __restrict__ out)
{
    const int c = blockIdx.x * blockDim.x + threadIdx.x;
    if (c >= MID) return;
    float sum = 0.f, sq = 0.f;
    for (int b = 0; b < BATCH; ++b) {
        const float v = h[b * MID + c];
        sum += v; sq += v * v;
    }
    const float mean = sum * (1.f / BATCH);
    const float var  = sq * (1.f / BATCH) - mean * mean;
    const float rstd = rsqrtf(var + 1e-5f);
    const float g = gamma[c], be = beta[c];
    for (int b = 0; b < BATCH; ++b) {
        const float v = (h[b * MID + c] - mean) * rstd * g + be;
        out[b * MID + c] = fmaxf(v, 0.f);
    }
}

// ---------------------------------------------------------------------------
extern "C" void kernel_launch(void* const* d_in, const int* in_sizes, int n_in,
                              void* d_out, int out_size, void* d_ws,
                              size_t ws_size, hipStream_t stream)
{
    const float* features  = (const float*)d_in[0];  // (32,512,7,500)
    const float* conv_w    = (const float*)d_in[1];  // (512,512,7,1) -> 512x3584
    const float* conv_b    = (const float*)d_in[2];  // (512,)
    const float* cc_w      = (const float*)d_in[3];  // (10,512,7,1)  -> 10x3584
    const float* cc_b      = (const float*)d_in[4];  // (10,)
    const float* centroids = (const float*)d_in[5];  // (10,512)
    const float* fc_w      = (const float*)d_in[6];  // (4096,512)
    const float* fc_b      = (const float*)d_in[7];  // (512,)
    const float* bn_gamma  = (const float*)d_in[8];  // (512,)
    const float* bn_beta   = (const float*)d_in[9];  // (512,)
    const float* logit_w   = (const float*)d_in[10]; // (512,5994)

    float* out_emb  = (float*)d_out;                 // (32,512)
    float* out_vlad = out_emb + BATCH * MID;         // (32,5994)

    // workspace layout (floats): ~35.3 MB total
    float* ws   = (float*)d_ws;
    float* feat = ws;                                 // 32*512*500
    float* scr  = feat + (long)BATCH * MID * WIDTH;   // 32*10*500
    float* soft = scr + (long)BATCH * NCLU * WIDTH;   // 32*10*500
    float* sa   = soft + (long)BATCH * NCLU * WIDTH;  // 32*10
    float* agg  = sa + BATCH * NCLU;                  // 32*10*512
    float* emb0 = agg + (long)BATCH * NCLU * MID;     // 32*4096
    float* h    = emb0 + (long)BATCH * KCLU * MID;    // 32*512
    (void)ws_size; (void)in_sizes; (void)n_in; (void)out_size;

    const dim3 blk(256);
    const long featStride = (long)MID * 7 * WIDTH;    // per-batch features

    // 1) feat = relu(conv_w(512x3584) @ X_b(3584x500) + conv_b)   [dominant]
    gemm_wmma_f32<false, 1, true><<<dim3(4, 8, BATCH), blk, 0, stream>>>(
        conv_w, features, conv_b, feat,
        MID, WIDTH, KDIM, KDIM, WIDTH, WIDTH,
        0L, featStride, (long)MID * WIDTH);

    // 2) score = cc_w(10x3584) @ X_b + cc_b
    gemm_wmma_f32<false, 1, false><<<dim3(4, 1, BATCH), blk, 0, stream>>>(
        cc_w, features, cc_b, scr,
        NCLU, WIDTH, KDIM, KDIM, WIDTH, WIDTH,
        0L, featStride, (long)NCLU * WIDTH);

    // 3) softmax over clusters, per (b,w)
    softmax_k<<<dim3((BATCH * WIDTH + 255) / 256), blk, 0, stream>>>(scr, soft);

    // 4) sa_sum
    sasum_k<<<dim3((BATCH * NCLU + 255) / 256), blk, 0, stream>>>(soft, sa);

    // 5) agg_b(10x512) = soft_b(10x500) @ feat_b^T(500x512)   (TRANS_B)
    gemm_wmma_f32<true, 0, false><<<dim3(4, 1, BATCH), blk, 0, stream>>>(
        soft, feat, nullptr, agg,
        NCLU, MID, WIDTH, WIDTH, WIDTH, MID,
        (long)NCLU * WIDTH, (long)MID * WIDTH, (long)NCLU * MID);

    // 6) residual + intra-norm -> emb0 (32x4096)
    cluster_k<<<dim3(BATCH * KCLU), blk, 0, stream>>>(agg, sa, centroids, emb0);

    // 7) h = emb0(32x4096) @ fc_w(4096x512) + fc_b (per-N bias)
    gemm_wmma_f32<false, 2, false><<<dim3(4, 1, 1), blk, 0, stream>>>(
        emb0, fc_w, fc_b, h,
        BATCH, MID, KCLU * MID, KCLU * MID, MID, MID, 0L, 0L, 0L);

    // 8) batch-norm + relu -> embeddings (output 0)
    bn_relu_k<<<dim3(2), blk, 0, stream>>>(h, bn_gamma, bn_beta, out_emb);

    // 9) vlad = embeddings(32x512) @ logit_w(512x5994) (output 1)
    gemm_wmma_f32<false, 0, false><<<dim3(47, 1, 1), blk, 0, stream>>>(
        out_emb, logit_w, nullptr, out_vlad,
        BATCH, OUTD, MID, MID, OUTD, OUTD, 0L, 0L, 0L);
}